// SpatialFilter_39118562132365
// MI455X (gfx1250) — compile-verified
//
#include <hip/hip_runtime.h>
#include <hip/hip_bf16.h>
#include <math.h>

// Permutohedral filter for PD=3, VD=5 (4 channels + 1 homogeneous), N = 64*128*128.
// splat (WMMA elevation + hash insert + atomic adds) -> 4 compacted blur passes -> slice.

#define HB    1024
#define HOFF  512
#define TBITS 20
#define TSIZE (1 << TBITS)          // 1M hash slots (expected occupancy ~1e4..1e5)
#define TMASK (TSIZE - 1)
#define HEMPTY (-1)
#define VD 5
#define SLOTCAP (1 << 18)           // compacted occupied-slot list capacity

typedef __attribute__((ext_vector_type(2))) float v2f;
typedef __attribute__((ext_vector_type(8))) float v8f;

// Elevation matrix E (4x3), rows scaled per reference _elev_mat(). Padded K to 4.
__constant__ float c_EM[16] = {
   2.3094010767585034f,  1.3333333333333333f,  0.9428090415820634f, 0.f,
  -2.3094010767585034f,  1.3333333333333333f,  0.9428090415820634f, 0.f,
   0.f,                 -2.6666666666666665f,  0.9428090415820634f, 0.f,
   0.f,                  0.f,                 -2.8284271247461903f, 0.f,
};

__device__ __forceinline__ unsigned hpos(int h) {
  return ((unsigned)h * 2654435761u) & TMASK;
}

// Insert; the CAS winner appends the slot to the compacted list.
__device__ __forceinline__ int hash_insert(int* keys, int* slotlist, int* count, int h) {
  unsigned pos = hpos(h);
  for (;;) {
    int prev = atomicCAS(&keys[pos], HEMPTY, h);
    if (prev == HEMPTY) {
      int idx = atomicAdd(count, 1);
      if (idx < SLOTCAP) slotlist[idx] = (int)pos;
      return (int)pos;
    }
    if (prev == h) return (int)pos;
    pos = (pos + 1) & TMASK;
  }
}

__device__ __forceinline__ int hash_find(const int* __restrict__ keys, int h) {
  if (h == HEMPTY) return -1;   // never alias the empty sentinel
  unsigned pos = hpos(h);
  for (int it = 0; it < TSIZE; ++it) {
    int k = keys[pos];
    if (k == h) return (int)pos;
    if (k == HEMPTY) return -1;
    pos = (pos + 1) & TMASK;
  }
  return -1;
}

__global__ void k_init(int* __restrict__ keys, float* __restrict__ lat0,
                       int* __restrict__ count) {
  int s = blockIdx.x * blockDim.x + threadIdx.x;
  if (s == 0) *count = 0;
  if (s < TSIZE) {
    keys[s] = HEMPTY;
    float* p = lat0 + (size_t)s * VD;
    #pragma unroll
    for (int c = 0; c < VD; ++c) p[c] = 0.f;
  }
}

__global__ __launch_bounds__(256) void k_splat(
    const float* __restrict__ input, const float* __restrict__ vg,
    const float* __restrict__ sg,
    int* __restrict__ keys, float* __restrict__ lat0,
    int* __restrict__ slotlist, int* __restrict__ count,
    int* __restrict__ sidx, float* __restrict__ sbary,
    int N, int H, int W)
{
  __shared__ float sf[256][4];     // features, K padded to 4
  __shared__ float selev[256][4];  // elevated coords
  const int tid = threadIdx.x;
  const int n = blockIdx.x * 256 + tid;   // grid exactly covers N -> no divergence

  // ---- features from voxel coordinates (reference order: [zz, yy, xx]) ----
  const int hw = H * W;
  const int z = n / hw;
  const int rmod = n - z * hw;
  const int y = rmod / W;
  const int x = rmod - y * W;
  sf[tid][0] = vg[0] * (float)z / sg[0];
  sf[tid][1] = vg[2] * (float)y / sg[2];
  sf[tid][2] = vg[1] * (float)x / sg[1];
  sf[tid][3] = 0.f;
  __syncthreads();

  // ---- WMMA elevation: D(16x16) = A(16x4 feats) x B(4x16 = E^T padded) ----
  const int lane  = tid & 31;
  const int wbase = tid & ~31;     // wave's first point in block
  const int m     = lane & 15;     // A row / B,D column index
  const int half  = lane >> 4;     // lane half select
  const int kh    = half * 2;      // K base for this half (A & B layouts)

  v2f B;
  B.x = (m < 4) ? c_EM[m * 4 + kh]     : 0.f;   // B[k][n] = E[n][k]
  B.y = (m < 4) ? c_EM[m * 4 + kh + 1] : 0.f;

  v2f A0, A1;
  A0.x = sf[wbase      + m][kh];  A0.y = sf[wbase      + m][kh + 1];
  A1.x = sf[wbase + 16 + m][kh];  A1.y = sf[wbase + 16 + m][kh + 1];

  v8f Cz = {};
  v8f D0 = __builtin_amdgcn_wmma_f32_16x16x4_f32(false, A0, false, B, (short)0, Cz, false, false);
  v8f D1 = __builtin_amdgcn_wmma_f32_16x16x4_f32(false, A1, false, B, (short)0, Cz, false, false);

  if (m < 4) {  // only columns n<4 carry elevated dims
    #pragma unroll
    for (int j = 0; j < 8; ++j) {
      selev[wbase      + half * 8 + j][m] = D0[j];
      selev[wbase + 16 + half * 8 + j][m] = D1[j];
    }
  }
  __syncthreads();

  float e[4];
  #pragma unroll
  for (int i = 0; i < 4; ++i) e[i] = selev[tid][i];

  // ---- enclosing simplex: rem0, rank ----
  float rem0[4], d[4];
  int rank[4];
  float sumrd = 0.f;
  #pragma unroll
  for (int i = 0; i < 4; ++i) {
    float v  = e[i] * 0.25f;
    float up = ceilf(v), dn = floorf(v);
    float rd = ((up - v) < (v - dn)) ? up : dn;
    rem0[i]  = rd * 4.0f;
    d[i]     = e[i] - rem0[i];
    sumrd   += rd;
  }
  #pragma unroll
  for (int i = 0; i < 4; ++i) {
    int rr = 0;
    #pragma unroll
    for (int j = 0; j < 4; ++j) {
      if (j > i)      rr += (d[i] <  d[j]);
      else if (j < i) rr += (d[i] <= d[j]);
    }
    rank[i] = rr;
  }
  const int sums = (int)rintf(sumrd);
  #pragma unroll
  for (int i = 0; i < 4; ++i) {
    rank[i] += sums;
    if (rank[i] < 0)      { rank[i] += 4; rem0[i] += 4.f; }
    else if (rank[i] > 3) { rank[i] -= 4; rem0[i] -= 4.f; }
  }

  // ---- barycentric weights ----
  float bb[5] = {0.f, 0.f, 0.f, 0.f, 0.f};
  #pragma unroll
  for (int i = 0; i < 4; ++i) {
    float t = (e[i] - rem0[i]) * 0.25f;
    bb[3 - rank[i]] += t;
    bb[4 - rank[i]] -= t;
  }
  bb[0] += 1.0f + bb[4];

  // ---- splat the 4 simplex vertices ----
  const int k0 = (int)rem0[0], k1 = (int)rem0[1], k2 = (int)rem0[2];
  const float v0 = input[0 * (size_t)N + n];
  const float v1 = input[1 * (size_t)N + n];
  const float v2 = input[2 * (size_t)N + n];
  const float v3 = input[3 * (size_t)N + n];
  int slots[4];
  #pragma unroll
  for (int r = 0; r < 4; ++r) {
    const int c0 = k0 + ((rank[0] >= 4 - r) ? r - 4 : r);
    const int c1 = k1 + ((rank[1] >= 4 - r) ? r - 4 : r);
    const int c2 = k2 + ((rank[2] >= 4 - r) ? r - 4 : r);
    const int h  = ((c0 + HOFF) * HB + (c1 + HOFF)) * HB + (c2 + HOFF);
    const int slot = hash_insert(keys, slotlist, count, h);
    slots[r] = slot;
    const float w = bb[r];
    float* lp = lat0 + (size_t)slot * VD;
    atomicAdd(lp + 0, v0 * w);
    atomicAdd(lp + 1, v1 * w);
    atomicAdd(lp + 2, v2 * w);
    atomicAdd(lp + 3, v3 * w);
    atomicAdd(lp + 4, w);
  }
  // vectorized b128 stores of per-point slot indices and barycentrics
  *(int4*)  (sidx  + (size_t)n * 4) = make_int4(slots[0], slots[1], slots[2], slots[3]);
  *(float4*)(sbary + (size_t)n * 4) = make_float4(bb[0], bb[1], bb[2], bb[3]);
}

// Blur only over the compacted occupied-slot list.
__global__ void k_blur(const int* __restrict__ keys,
                       const int* __restrict__ slotlist,
                       const int* __restrict__ count,
                       const float* __restrict__ src,
                       float* __restrict__ dst, int delta)
{
  int i = blockIdx.x * blockDim.x + threadIdx.x;
  int cnt = *count;
  if (cnt > SLOTCAP) cnt = SLOTCAP;
  if (i >= cnt) return;
  const int s = slotlist[i];
  const int h = keys[s];

  const float* cp = src + (size_t)s * VD;
  float o0 = cp[0], o1 = cp[1], o2 = cp[2], o3 = cp[3], o4 = cp[4];
  int p = hash_find(keys, h + delta);
  if (p >= 0) {
    const float* q = src + (size_t)p * VD;
    o0 += 0.5f * q[0]; o1 += 0.5f * q[1]; o2 += 0.5f * q[2];
    o3 += 0.5f * q[3]; o4 += 0.5f * q[4];
  }
  p = hash_find(keys, h - delta);
  if (p >= 0) {
    const float* q = src + (size_t)p * VD;
    o0 += 0.5f * q[0]; o1 += 0.5f * q[1]; o2 += 0.5f * q[2];
    o3 += 0.5f * q[3]; o4 += 0.5f * q[4];
  }
  float* dp = dst + (size_t)s * VD;
  dp[0] = o0; dp[1] = o1; dp[2] = o2; dp[3] = o3; dp[4] = o4;
}

__global__ void k_slice(const float* __restrict__ lat,
                        const int* __restrict__ sidx,
                        const float* __restrict__ sbary,
                        float* __restrict__ out, int N)
{
  int n = blockIdx.x * blockDim.x + threadIdx.x;
  if (n >= N) return;
  const int4   s4 = *(const int4*)  (sidx  + (size_t)n * 4);
  const float4 w4 = *(const float4*)(sbary + (size_t)n * 4);
  const int   ss[4] = {s4.x, s4.y, s4.z, s4.w};
  const float ww[4] = {w4.x, w4.y, w4.z, w4.w};
  float a0 = 0.f, a1 = 0.f, a2 = 0.f, a3 = 0.f, a4 = 0.f;
  #pragma unroll
  for (int r = 0; r < 4; ++r) {
    const float* lp = lat + (size_t)ss[r] * VD;
    const float w = ww[r];
    a0 += w * lp[0]; a1 += w * lp[1]; a2 += w * lp[2];
    a3 += w * lp[3]; a4 += w * lp[4];
  }
  const float alpha = 0.8888888888888889f;   // 1/(1+2^-3)
  const float norm  = alpha * a4;
  const float inv   = 1.0f / (norm + 2.2204460492503131e-16f);
  out[0 * (size_t)N + n] = alpha * a0 * inv;
  out[1 * (size_t)N + n] = alpha * a1 * inv;
  out[2 * (size_t)N + n] = alpha * a2 * inv;
  out[3 * (size_t)N + n] = alpha * a3 * inv;
}

extern "C" void kernel_launch(void* const* d_in, const int* in_sizes, int n_in,
                              void* d_out, int out_size, void* d_ws, size_t ws_size,
                              hipStream_t stream) {
  const float* input = (const float*)d_in[0];
  // d_in[1] (image) only supplies shape in the reference; values unused.
  const float* vg = (const float*)d_in[2];
  const float* sg = (const float*)d_in[3];

  const int Dd = 64, Hh = 128, Ww = 128;
  const int N  = Dd * Hh * Ww;               // 1,048,576

  // workspace layout
  char* ws = (char*)d_ws;
  int*   keys     = (int*)  (ws);                                     //  4 * TSIZE
  float* lat0     = (float*)(ws +  4ull * TSIZE);                     // 20 * TSIZE
  float* lat1     = (float*)(ws + 24ull * TSIZE);                     // 20 * TSIZE
  int*   sidx     = (int*)  (ws + 44ull * TSIZE);                     // 16 * N
  float* sbary    = (float*)(ws + 44ull * TSIZE + 16ull * (size_t)N); // 16 * N
  int*   slotlist = (int*)  (ws + 44ull * TSIZE + 32ull * (size_t)N); //  4 * SLOTCAP
  int*   count    = (int*)  (ws + 44ull * TSIZE + 32ull * (size_t)N + 4ull * SLOTCAP);

  const int threads = 256;
  const int gridT = TSIZE / threads;
  const int gridN = N / threads;
  const int gridB = SLOTCAP / threads;

  k_init<<<gridT, threads, 0, stream>>>(keys, lat0, count);

  k_splat<<<gridN, threads, 0, stream>>>(input, vg, sg, keys, lat0,
                                         slotlist, count, sidx, sbary, N, Hh, Ww);

  // blur deltas: hash(k + o) = h + (o0*2^20 + o1*2^10 + o2)
  const int deltas[4] = {
    -3 * 1048576 + 1024 + 1,   // o = (-3, 1, 1)
     1048576 - 3 * 1024 + 1,   // o = ( 1,-3, 1)
     1048576 + 1024 - 3,       // o = ( 1, 1,-3)
     1048576 + 1024 + 1        // o = ( 1, 1, 1)
  };
  k_blur<<<gridB, threads, 0, stream>>>(keys, slotlist, count, lat0, lat1, deltas[0]);
  k_blur<<<gridB, threads, 0, stream>>>(keys, slotlist, count, lat1, lat0, deltas[1]);
  k_blur<<<gridB, threads, 0, stream>>>(keys, slotlist, count, lat0, lat1, deltas[2]);
  k_blur<<<gridB, threads, 0, stream>>>(keys, slotlist, count, lat1, lat0, deltas[3]);

  k_slice<<<gridN, threads, 0, stream>>>(lat0, sidx, sbary, (float*)d_out, N);
}